// TransformerConv_45561013076143
// MI455X (gfx1250) — compile-verified
//
#include <hip/hip_runtime.h>
#include <hip/hip_bf16.h>
#include <math.h>

// ---------------------------------------------------------------------------
// TransformerConv for MI455X (gfx1250).
//  N=4096 nodes, IN=128, H=4 heads, C=32 (HC=128).
//  Phase 1: Q/K/V/Xr projections via V_WMMA_F32_16X16X4_F32 (exact fp32).
//  Phase 2: fused masked flash-attention, 1 wave per (16-row tile, head).
//           adj (64MB) is the bandwidth floor; it fits in the 192MB L2 so
//           the 4-head re-reads are L2 hits. attn [4,4096,4096] is never
//           materialized.
//  Phase 3: beta gating (sigmoid of 384-dot) + skip mix.
// Workspace: 5 x [4096,128] f32 = 10 MB in d_ws.
// ---------------------------------------------------------------------------

typedef float v2f __attribute__((ext_vector_type(2)));
typedef float v8f __attribute__((ext_vector_type(8)));

#define NND   4096
#define HCDIM 128
#define CDIM  32
#define HDIM  4

static __device__ __forceinline__ v8f wmma_f32(v2f a, v2f b, v8f c) {
  return __builtin_amdgcn_wmma_f32_16x16x4_f32(
      /*neg_a=*/false, a, /*neg_b=*/false, b,
      /*c_mod=*/(short)0, c, /*reuse_a=*/false, /*reuse_b=*/false);
}

// ---------------------------------------------------------------------------
// Kernel 1: out[m, n0:n0+16] = x[m,:] @ W[:, n0:n0+16] + bias   (M=4096,K=128)
// One wave per 16x16 output tile, 32 k-steps of 16x16x4 f32 WMMA.
// ---------------------------------------------------------------------------
__global__ void __launch_bounds__(32)
gemm128_bias(const float* __restrict__ x, const float* __restrict__ W,
             const float* __restrict__ bias, float* __restrict__ out) {
  const int lane = threadIdx.x;
  const int half = lane >> 4;      // 0: lanes 0-15, 1: lanes 16-31
  const int l16  = lane & 15;
  const int m0 = blockIdx.x * 16;
  const int n0 = blockIdx.y * 16;

  v8f acc = {};
  const float* xrow = x + (size_t)(m0 + l16) * HCDIM + 2 * half;
  const float* wcol = W + n0 + l16;
#pragma unroll
  for (int st = 0; st < 32; ++st) {
    // A: lane=m, elems k = 4*st + 2*half + {0,1}  -> one b64 load
    v2f a = *(const v2f*)(xrow + 4 * st);
    // B: lane=n, elems k = 4*st + 2*half + {0,1}
    const int k0 = 4 * st + 2 * half;
    v2f b = { wcol[(size_t)k0 * HCDIM], wcol[(size_t)(k0 + 1) * HCDIM] };
    acc = wmma_f32(a, b, acc);
  }
  const float bv = bias[n0 + l16];
#pragma unroll
  for (int r = 0; r < 8; ++r) {
    const int m = m0 + r + 8 * half;           // C/D layout: vgpr r -> row r / r+8
    out[(size_t)m * HCDIM + n0 + l16] = acc[r] + bv;
  }
}

// ---------------------------------------------------------------------------
// Kernel 2: fused masked attention, flash-style.
//  grid = (256 row-tiles, 4 heads), block = 32 (one wave).
//  S = Q Kt (8 WMMAs), mask from adj, online softmax, O += P V (8 WMMAs).
//  Output written in the reference's permuted layout: col = c*H + h.
// ---------------------------------------------------------------------------
__global__ void __launch_bounds__(32)
attn_flash(const float* __restrict__ Q, const float* __restrict__ K,
           const float* __restrict__ V, const float* __restrict__ adj,
           float* __restrict__ out) {
  const int lane = threadIdx.x;
  const int half = lane >> 4;
  const int l16  = lane & 15;
  const int h  = blockIdx.y;
  const int m0 = blockIdx.x * 16;

  __shared__ float pbuf[16 * 20];   // stride 20 to dodge bank conflicts

  // Q tile A-operands: 8 steps over C=32, kept in registers for all 256 iters
  v2f qa[8];
  {
    const float* qrow = Q + (size_t)(m0 + l16) * HCDIM + h * CDIM + 2 * half;
#pragma unroll
    for (int st = 0; st < 8; ++st) qa[st] = *(const v2f*)(qrow + 4 * st);
  }

  float mrun[8], lrun[8];
  v8f o0 = {}, o1 = {};               // O accumulator: channels 0-15 / 16-31
#pragma unroll
  for (int r = 0; r < 8; ++r) { mrun[r] = -INFINITY; lrun[r] = 0.0f; }

  const float inv_sqrt_c = 0.17677669529663687f;   // 1/sqrt(32)

  for (int n0 = 0; n0 < NND; n0 += 16) {
    // ---- S = Q * K^T  (contract over C=32) ----
    v8f s = {};
    const float* krow = K + (size_t)(n0 + l16) * HCDIM + h * CDIM + 2 * half;
#pragma unroll
    for (int st = 0; st < 8; ++st) {
      v2f kb = *(const v2f*)(krow + 4 * st);   // B: lane=n, k=4st+2half+{0,1}
      s = wmma_f32(qa[st], kb, s);
    }

    // ---- mask + online softmax (row m lives in one 16-lane half per vgpr) --
    const float* arow = adj + (size_t)(m0 + 8 * half) * NND + n0 + l16;
    float p[8];
#pragma unroll
    for (int r = 0; r < 8; ++r) {
      const float av  = arow[(size_t)r * NND];
      const bool  msk = (av == 0.0f);
      const float sv  = msk ? -INFINITY : (s[r] * inv_sqrt_c);
      // row max over the 16 columns of this tile (lanes within a half)
      float mx = sv;
      mx = fmaxf(mx, __shfl_xor(mx, 1));
      mx = fmaxf(mx, __shfl_xor(mx, 2));
      mx = fmaxf(mx, __shfl_xor(mx, 4));
      mx = fmaxf(mx, __shfl_xor(mx, 8));
      const float mnew = fmaxf(mrun[r], mx);
      const float scale = (mnew == mrun[r]) ? 1.0f : __expf(mrun[r] - mnew);
      const float msafe = (mnew == -INFINITY) ? 0.0f : mnew;
      const float e = msk ? 0.0f : __expf(sv - msafe);
      float rs = e;
      rs += __shfl_xor(rs, 1);
      rs += __shfl_xor(rs, 2);
      rs += __shfl_xor(rs, 4);
      rs += __shfl_xor(rs, 8);
      lrun[r] = lrun[r] * scale + rs;
      mrun[r] = mnew;
      o0[r] *= scale;
      o1[r] *= scale;
      p[r] = e;
    }

    // ---- relayout P: C/D layout -> A layout via wave-private LDS ----
#pragma unroll
    for (int r = 0; r < 8; ++r)
      pbuf[(r + 8 * half) * 20 + l16] = p[r];          // pbuf[m*20 + n]
    asm volatile("s_wait_dscnt 0" ::: "memory");       // cross-lane LDS dep

    // ---- O += P * V  (contract over the 16 tile columns) ----
    const float* vrow = V + (size_t)n0 * HCDIM + h * CDIM + l16;
#pragma unroll
    for (int st = 0; st < 4; ++st) {
      const int k0 = 4 * st + 2 * half;
      v2f pa = *(const v2f*)(&pbuf[l16 * 20 + k0]);    // A: lane=m, k=n
      const float* vb = vrow + (size_t)k0 * HCDIM;
      v2f vlo = { vb[0],  vb[HCDIM] };                 // B: c = l16
      v2f vhi = { vb[16], vb[HCDIM + 16] };            // B: c = l16+16
      o0 = wmma_f32(pa, vlo, o0);
      o1 = wmma_f32(pa, vhi, o1);
    }
  }

  // ---- finalize: divide by row sum (isolated rows -> 0), permuted store ----
#pragma unroll
  for (int r = 0; r < 8; ++r) {
    const float inv = (lrun[r] == 0.0f) ? 0.0f : (1.0f / lrun[r]);
    const int m = m0 + r + 8 * half;
    out[(size_t)m * HCDIM + l16 * HDIM + h]        = o0[r] * inv;  // c = l16
    out[(size_t)m * HCDIM + (l16 + 16) * HDIM + h] = o1[r] * inv;  // c = l16+16
  }
}

// ---------------------------------------------------------------------------
// Kernel 3: beta = sigmoid([out, x_r, out-x_r] @ Wb); y = b*x_r + (1-b)*out
// One 128-thread block per node.
// ---------------------------------------------------------------------------
__global__ void __launch_bounds__(128)
beta_gate(const float* __restrict__ ao, const float* __restrict__ xr,
          const float* __restrict__ Wb, float* __restrict__ out) {
  const int n = blockIdx.x;
  const int j = threadIdx.x;
  __shared__ float red[128];

  const float o = ao[(size_t)n * HCDIM + j];
  const float r = xr[(size_t)n * HCDIM + j];
  red[j] = o * Wb[j] + r * Wb[HCDIM + j] + (o - r) * Wb[2 * HCDIM + j];
  __syncthreads();
#pragma unroll
  for (int s = 64; s > 0; s >>= 1) {
    if (j < s) red[j] += red[j + s];
    __syncthreads();
  }
  const float beta = 1.0f / (1.0f + __expf(-red[0]));
  out[(size_t)n * HCDIM + j] = beta * r + (1.0f - beta) * o;
}

// ---------------------------------------------------------------------------
extern "C" void kernel_launch(void* const* d_in, const int* in_sizes, int n_in,
                              void* d_out, int out_size, void* d_ws, size_t ws_size,
                              hipStream_t stream) {
  (void)in_sizes; (void)n_in; (void)out_size; (void)ws_size;
  const float* x   = (const float*)d_in[0];
  const float* adj = (const float*)d_in[1];
  const float* Wq  = (const float*)d_in[2];
  const float* bq  = (const float*)d_in[3];
  const float* Wk  = (const float*)d_in[4];
  const float* bk  = (const float*)d_in[5];
  const float* Wv  = (const float*)d_in[6];
  const float* bv  = (const float*)d_in[7];
  const float* Ws  = (const float*)d_in[8];
  const float* bs  = (const float*)d_in[9];
  const float* Wb  = (const float*)d_in[10];
  float* outp = (float*)d_out;

  float* ws  = (float*)d_ws;
  const size_t MAT = (size_t)NND * HCDIM;   // 524288 floats
  float* Qm = ws;
  float* Km = ws + 1 * MAT;
  float* Vm = ws + 2 * MAT;
  float* Xr = ws + 3 * MAT;
  float* AO = ws + 4 * MAT;                 // attention output (permuted cols)

  dim3 g1(NND / 16, HCDIM / 16);
  gemm128_bias<<<g1, 32, 0, stream>>>(x, Wq, bq, Qm);
  gemm128_bias<<<g1, 32, 0, stream>>>(x, Wk, bk, Km);
  gemm128_bias<<<g1, 32, 0, stream>>>(x, Wv, bv, Vm);
  gemm128_bias<<<g1, 32, 0, stream>>>(x, Ws, bs, Xr);

  attn_flash<<<dim3(NND / 16, HDIM), 32, 0, stream>>>(Qm, Km, Vm, adj, AO);

  beta_gate<<<NND, 128, 0, stream>>>(AO, Xr, Wb, outp);
}